// BilinearInteraction_38817914421731
// MI455X (gfx1250) — compile-verified
//
#include <hip/hip_runtime.h>

// Problem constants (match the reference)
#define FIELD   26
#define EMB     32
#define NPAIRS  325          // C(26,2)
#define MTILE   16           // batch rows per wave-tile
#define PAIRS_PER_BLOCK 8    // one pair per wave, 8 waves per block
#define XROW    (FIELD*EMB)          // 832 floats per batch row of x
#define LDSROW  (XROW + 4)           // 836: pad so 16-lane column reads hit distinct banks

typedef __attribute__((ext_vector_type(2))) float v2f;
typedef __attribute__((ext_vector_type(8))) float v8f;

__global__ __launch_bounds__(256)
void bilinear_interaction_wmma(const float* __restrict__ x,
                               const float* __restrict__ W,
                               const float* __restrict__ bias,
                               float* __restrict__ out)
{
    __shared__ float xs[MTILE * LDSROW];   // 53,504 bytes

    const int m0 = blockIdx.x * MTILE;     // first batch row of this tile

    // ---- Stage x[m0:m0+16, :, :] into LDS (float4 cooperative copy) ----
    {
        const int nvec = MTILE * (XROW / 4);          // 3328 float4
        for (int idx = threadIdx.x; idx < nvec; idx += blockDim.x) {
            const int row = idx / (XROW / 4);
            const int col = (idx - row * (XROW / 4)) * 4;
            const float4 v = *reinterpret_cast<const float4*>(
                x + (size_t)(m0 + row) * XROW + col);
            float* d = &xs[row * LDSROW + col];
            d[0] = v.x; d[1] = v.y; d[2] = v.z; d[3] = v.w;
        }
    }
    __syncthreads();

    const int wave = threadIdx.x >> 5;
    const int lane = threadIdx.x & 31;
    const int p    = blockIdx.y * PAIRS_PER_BLOCK + wave;
    if (p >= NPAIRS) return;

    // ---- Decode triu pair index p -> (i, j), wave-uniform (SALU loop) ----
    int i = 0, rem = p, cnt = FIELD - 1;
    while (rem >= cnt) { rem -= cnt; ++i; --cnt; }
    const int j = i + 1 + rem;

    const int hi = lane >> 4;        // 0: K0/K1 & M-low half, 1: K2/K3 & M-high half
    const int lm = lane & 15;        // M (for A) / N==e (for B, C, D)

    // ---- 16x32 output tile: two 16x16 WMMA accumulators over K=32 (8 x K=4) ----
    v8f acc0 = {};                   // e in [0,16)
    v8f acc1 = {};                   // e in [16,32)

    const float* Wp  = W + (size_t)p * (EMB * EMB);          // W[p][e][d], e-major
    const float* a_base  = &xs[lm * LDSROW + i * EMB + 2 * hi];
    const float* b0_base = Wp + (size_t)lm        * EMB + 2 * hi;
    const float* b1_base = Wp + (size_t)(lm + 16) * EMB + 2 * hi;

#pragma unroll
    for (int kk = 0; kk < 8; ++kk) {
        const int kb = kk * 4;
        const v2f a  = *reinterpret_cast<const v2f*>(a_base  + kb); // A: 16x4 f32
        const v2f b0 = *reinterpret_cast<const v2f*>(b0_base + kb); // B: 4x16 f32 (= W^T tile 0)
        const v2f b1 = *reinterpret_cast<const v2f*>(b1_base + kb); // B tile 1
        acc0 = __builtin_amdgcn_wmma_f32_16x16x4_f32(
                   false, a, false, b0, (short)0, acc0, false, false);
        acc1 = __builtin_amdgcn_wmma_f32_16x16x4_f32(
                   false, a, false, b1, (short)0, acc1, false, false);
    }

    // ---- Epilogue: (+bias) * v_j, store ----
    const float bias0 = bias[p * EMB + lm];
    const float bias1 = bias[p * EMB + 16 + lm];
    float* outp = out + (size_t)p * EMB;

#pragma unroll
    for (int r = 0; r < 8; ++r) {
        const int row = r + 8 * hi;                 // batch row within tile (C/D layout)
        const float vj0 = xs[row * LDSROW + j * EMB + lm];
        const float vj1 = xs[row * LDSROW + j * EMB + 16 + lm];
        float* o = outp + (size_t)(m0 + row) * (NPAIRS * EMB);
        o[lm]      = (acc0[r] + bias0) * vj0;
        o[16 + lm] = (acc1[r] + bias1) * vj1;
    }
}

extern "C" void kernel_launch(void* const* d_in, const int* in_sizes, int n_in,
                              void* d_out, int out_size, void* d_ws, size_t ws_size,
                              hipStream_t stream) {
    (void)in_sizes; (void)n_in; (void)d_ws; (void)ws_size; (void)out_size;
    const float* x    = (const float*)d_in[0];   // (4096, 26, 32)  f32
    const float* W    = (const float*)d_in[1];   // (325, 32, 32)   f32
    const float* bias = (const float*)d_in[2];   // (325, 32)       f32
    float* out = (float*)d_out;                  // (4096, 325, 32) f32

    const int B = 4096;
    dim3 grid(B / MTILE, (NPAIRS + PAIRS_PER_BLOCK - 1) / PAIRS_PER_BLOCK);
    dim3 block(256);
    bilinear_interaction_wmma<<<grid, block, 0, stream>>>(x, W, bias, out);
}